// ImageGS_27676769255705
// MI455X (gfx1250) — compile-verified
//
#include <hip/hip_runtime.h>

// ---------------------------------------------------------------------------
// ImageGS: render 16384 pixels against 4096 rotated/scaled 2D Gaussians,
// top-8 weighted color blend.
//
// Strategy (MI455X / gfx1250, wave32):
//   exponent(p,n) = -0.5*(rx^2+ry^2) expands to a dot product between an
//   8-dim pixel feature vector  f(p) = {cx^2, cy^2, cx*cy, cx, cy, 1, 0, 0}
//   and a per-gaussian coefficient vector g(n) (precomputed by gs_coeff).
//   => P x N exponent matrix is a (P x 8) @ (8 x N) GEMM, evaluated with two
//   chained V_WMMA_F32_16X16X4_F32 per 16x16 tile (fp32 WMMA: the quadratic
//   expansion cancels terms up to ~2^14, so sub-fp32 A/B would be garbage).
//
//   A (SRC0) = gaussian coeffs (M = gaussian), B (SRC1) = pixel features
//   (N = pixel). Per the CDNA5 C/D layout, lane l owns pixel l%16 and
//   receives 8 exponents per chunk for gaussians mb + 8*(l>>4) + v -> the
//   running top-8 heap is private per lane (registers only). The expensive
//   scan-insert runs only when a candidate beats the cached heap-min
//   (execz-skipped fast path); the final merge of the two half-wave heaps
//   for the same pixel uses ds_bpermute.
//
//   Coefficients are stored K-permuted per row ({k0,k1,k4,k5, k2,k3,k6,k7})
//   so each lane's two v2f A fragments come from ONE aligned
//   global_load_b128 per 16-gaussian chunk.
// ---------------------------------------------------------------------------

typedef __attribute__((ext_vector_type(2))) float v2f;
typedef __attribute__((ext_vector_type(4))) float v4f;
typedef __attribute__((ext_vector_type(8))) float v8f;

#define NUMG 4096
#define NPIX 16384          // 4 * 4096 pixels
#define KTOP 8
#define INV_IMG 0.00390625f // 1/256
#define WAVES_PER_BLOCK 8   // 8 wave32 = 256 threads -> fills a WGP

// ---------------------------------------------------------------------------
// Kernel 1: per-gaussian quadratic coefficients, K-permuted:
//   row n = { q0, q1, q4, q5,   q2, q3, 0, 0 }
// where exponent = q0*cx^2 + q1*cy^2 + q2*cx*cy + q3*cx + q4*cy + q5
// (halves of the row = the b128 payloads for lane-half 0 / lane-half 1).
// ---------------------------------------------------------------------------
__global__ void __launch_bounds__(256) gs_coeff(const float* __restrict__ u,
                                                const float* __restrict__ t,
                                                const float* __restrict__ s,
                                                float* __restrict__ G) {
  int n = blockIdx.x * blockDim.x + threadIdx.x;
  if (n >= NUMG) return;
  float ux = u[2 * n + 0], uy = u[2 * n + 1];
  float ang = t[n];
  float s0 = s[2 * n + 0], s1 = s[2 * n + 1];
  float ct = __cosf(ang), st = __sinf(ang);
  float a1 = ct / s0, b1 = st / s0;
  float a2 = -st / s1, b2 = ct / s1;
  float c1 = -(a1 * ux + b1 * uy);
  float c2 = -(a2 * ux + b2 * uy);
  float q0 = -0.5f * (a1 * a1 + a2 * a2); // cx^2
  float q1 = -0.5f * (b1 * b1 + b2 * b2); // cy^2
  float q2 = -(a1 * b1 + a2 * b2);        // cx*cy
  float q3 = -(a1 * c1 + a2 * c2);        // cx
  float q4 = -(b1 * c1 + b2 * c2);        // cy
  float q5 = -0.5f * (c1 * c1 + c2 * c2); // 1
  v4f* g = (v4f*)(G + (size_t)n * 8);
  g[0] = (v4f){q0, q1, q4, q5}; // K = {0,1} and {4,5}  (lane-half 0)
  g[1] = (v4f){q2, q3, 0.0f, 0.0f}; // K = {2,3} and {6,7}  (lane-half 1)
}

// Branchless replace-min insertion into a register-resident top-K list.
__device__ __forceinline__ void topk_insert(float (&hv)[KTOP], int (&hi)[KTOP],
                                            float val, int id) {
  float mn = hv[0];
  int slot = 0;
#pragma unroll
  for (int j = 1; j < KTOP; ++j) {
    bool lt = hv[j] < mn;
    mn = lt ? hv[j] : mn;
    slot = lt ? j : slot;
  }
  bool ins = val > mn;
#pragma unroll
  for (int j = 0; j < KTOP; ++j) {
    bool w = ins & (j == slot);
    hv[j] = w ? val : hv[j];
    hi[j] = w ? id : hi[j];
  }
}

__device__ __forceinline__ float heap_min(const float (&hv)[KTOP]) {
  float m0 = fminf(hv[0], hv[1]);
  float m1 = fminf(hv[2], hv[3]);
  float m2 = fminf(hv[4], hv[5]);
  float m3 = fminf(hv[6], hv[7]);
  return fminf(fminf(m0, m1), fminf(m2, m3));
}

// ---------------------------------------------------------------------------
// Kernel 2: one wave32 per 16-pixel tile, 8 tiles per 256-thread block;
// 256 x (2 WMMA f32 16x16x4) per wave, one global_load_b128 per chunk.
// ---------------------------------------------------------------------------
__global__ void __launch_bounds__(256) gs_render(const float* __restrict__ x,
                                                 const float* __restrict__ G,
                                                 const float* __restrict__ c,
                                                 float* __restrict__ out) {
  const int lane = threadIdx.x & 31; // lane within wave32
  const int wave = threadIdx.x >> 5; // wave within block
  const int nsel = lane & 15; // pixel column (B/D N index) / gaussian row (A M)
  const int half = lane >> 4; // 0: K {0,1}/{4,5}, rows M 0..7 ; 1: K {2,3}/{6,7}, M 8..15
  const int pbase = (blockIdx.x * WAVES_PER_BLOCK + wave) * 16;
  const int kb = half * 2;

  // Pixel features for this wave's 16 pixels (loop-invariant B matrix).
  // Assumed f32 B layout (K x N = 4 x 16, 2 VGPRs), mirroring the A table:
  //   VGPR0: lanes 0-15 -> (K=0, N=lane), lanes 16-31 -> (K=2, N=lane-16)
  //   VGPR1: lanes 0-15 -> (K=1, N=lane), lanes 16-31 -> (K=3, N=lane-16)
  const int p = pbase + nsel;
  const float cx = x[2 * p + 0] * INV_IMG;
  const float cy = x[2 * p + 1] * INV_IMG;
  const float F[8] = {cx * cx, cy * cy, cx * cy, cx, cy, 1.0f, 0.0f, 0.0f};
  v2f B0 = {F[kb + 0], F[kb + 1]}; // K block 0..3
  v2f B1 = {F[kb + 4], F[kb + 5]}; // K block 4..7

  float hv[KTOP];
  int hi[KTOP];
#pragma unroll
  for (int j = 0; j < KTOP; ++j) {
    hv[j] = -3.0e38f;
    hi[j] = 0;
  }
  float hmin = -3.0e38f; // cached min of hv[]

  // Per-lane A pointer: row (mb + nsel), half-selected 16-byte sub-block.
  const v4f* ga = (const v4f*)(G + (size_t)nsel * 8 + (size_t)half * 4);

  for (int mb = 0; mb < NUMG; mb += 16) {
    // One b128 load supplies both v2f A fragments for this lane.
    const v4f a = ga[(size_t)mb * 2];
    v2f A0 = {a.x, a.y};
    v2f A1 = {a.z, a.w};
    v8f acc = {};
    acc = __builtin_amdgcn_wmma_f32_16x16x4_f32(false, A0, false, B0, (short)0,
                                                acc, false, false);
    acc = __builtin_amdgcn_wmma_f32_16x16x4_f32(false, A1, false, B1, (short)0,
                                                acc, false, false);
    // D[M][N]: lane (N = nsel) holds M = 8*half + v across the 8 acc regs.
    // Fast path: only run the scan-inserts if some candidate beats the
    // cached heap minimum (rare once the heap warms up). The divergent
    // region is fully re-converged before the next WMMA.
    bool any = false;
#pragma unroll
    for (int v = 0; v < 8; ++v) any |= (acc[v] > hmin);
    if (any) {
      const int idbase = mb + 8 * half;
#pragma unroll
      for (int v = 0; v < 8; ++v) topk_insert(hv, hi, acc[v], idbase + v);
      hmin = heap_min(hv);
    }
  }

  // Merge the hi-half lane's heap (same pixel, gaussian rows 8..15 of each
  // chunk) into the lo-half lane. Uniform bpermutes first (EXEC all-1s),
  // divergent insert afterwards.
  const int srcb = ((lane & 15) + 16) << 2; // byte index for ds_bpermute
  float ov[KTOP];
  int oi[KTOP];
#pragma unroll
  for (int j = 0; j < KTOP; ++j) {
    ov[j] = __int_as_float(
        __builtin_amdgcn_ds_bpermute(srcb, __float_as_int(hv[j])));
    oi[j] = __builtin_amdgcn_ds_bpermute(srcb, hi[j]);
  }

  if (half == 0) {
#pragma unroll
    for (int j = 0; j < KTOP; ++j) topk_insert(hv, hi, ov[j], oi[j]);

    float den = 1e-8f;
    float r = 0.0f, g = 0.0f, b = 0.0f;
#pragma unroll
    for (int j = 0; j < KTOP; ++j) {
      float w = __expf(hv[j]); // prob = exp(exponent); top-k(exp)==top-k(exponent)
      den += w;
      const float* cc = c + (size_t)hi[j] * 3;
      r += w * cc[0];
      g += w * cc[1];
      b += w * cc[2];
    }
    float inv = 1.0f / den;
    out[3 * (size_t)p + 0] = r * inv;
    out[3 * (size_t)p + 1] = g * inv;
    out[3 * (size_t)p + 2] = b * inv;
  }
}

// ---------------------------------------------------------------------------
extern "C" void kernel_launch(void* const* d_in, const int* in_sizes, int n_in,
                              void* d_out, int out_size, void* d_ws,
                              size_t ws_size, hipStream_t stream) {
  const float* x = (const float*)d_in[0]; // [4,4096,2]
  const float* u = (const float*)d_in[1]; // [4096,2]
  const float* t = (const float*)d_in[2]; // [4096]
  const float* s = (const float*)d_in[3]; // [4096,2]
  const float* c = (const float*)d_in[4]; // [4096,3]
  float* out = (float*)d_out;             // [4,4096,3]
  float* G = (float*)d_ws;                // [4096,8] permuted coeffs (128 KB)

  gs_coeff<<<(NUMG + 255) / 256, 256, 0, stream>>>(u, t, s, G);
  gs_render<<<NPIX / (16 * WAVES_PER_BLOCK), 256, 0, stream>>>(x, G, c, out);
}